// FSE_17995912970300
// MI455X (gfx1250) — compile-verified
//
#include <hip/hip_runtime.h>
#include <math.h>

typedef float v2f __attribute__((ext_vector_type(2)));
typedef float v8f __attribute__((ext_vector_type(8)));

#define N_BATCH   4096
#define D_FEAT    64
#define DET_EPS   1e-8f
#define NBLK      (N_BATCH / 16)   // 256 blocks of 16 rows
#define JTILES    4                // J-tiles per wave (register blocking)

// ---------------- Phase 1: elementwise precompute + row reductions ------------
// u  = var + mu^2 ; iv = 1/var ; mi = mu * iv
// s[row] = sum_d mu^2 * iv ; p[row] = prod_d var
__global__ void kl_prep_kernel(const float* __restrict__ mu,
                               const float* __restrict__ var,
                               float* __restrict__ u,
                               float* __restrict__ iv,
                               float* __restrict__ mi,
                               float* __restrict__ s,
                               float* __restrict__ p,
                               float* __restrict__ accum) {
  const int row  = blockIdx.x;          // 0..4095
  const int lane = threadIdx.x;         // 0..31 (wave32)
  if (row == 0 && lane == 0) { accum[0] = 0.0f; accum[1] = 0.0f; }

  float ssum = 0.0f, pprod = 1.0f;
#pragma unroll
  for (int t = 0; t < 2; ++t) {
    const int d   = lane + 32 * t;
    const int idx = row * D_FEAT + d;
    const float m   = mu[idx];
    const float v   = var[idx];
    const float inv = 1.0f / v;
    u [idx] = v + m * m;
    iv[idx] = inv;
    mi[idx] = m * inv;
    ssum  += m * m * inv;
    pprod *= v;
  }
#pragma unroll
  for (int off = 16; off > 0; off >>= 1) {
    ssum  += __shfl_xor(ssum,  off, 32);
    pprod *= __shfl_xor(pprod, off, 32);
  }
  if (lane == 0) { s[row] = ssum; p[row] = pprod; }
}

// ---------------- Phase 2: WMMA tile kernel, J-register-blocked ---------------
// One wave handles block-row bi against JTILES consecutive block-cols
// (bj0..bj0+3). A-side fragments are loaded once per k-step and reused against
// 4 B-fragment sets (loads/wmma: 2.0 -> 1.25). Both K(i,j) and K(j,i) tiles are
// produced in the SAME (m,n) register layout via operand-role swapping, so no
// register transpose is needed. fp32 WMMA (16x16x4) keeps the O(64)-magnitude
// trace terms exact enough for the KL differences.
__global__ __launch_bounds__(128, 1)
void kl_tile_kernel(const int*   __restrict__ labels,
                    const float* __restrict__ mu,
                    const float* __restrict__ u,
                    const float* __restrict__ iv,
                    const float* __restrict__ mi,
                    const float* __restrict__ s,
                    const float* __restrict__ p,
                    float* __restrict__ accum) {
  const int bi  = blockIdx.x;                       // 0..255
  const int jg  = blockIdx.y * 4 + threadIdx.y;     // 0..63, one wave each
  const int bj0 = jg * JTILES;                      // first J-tile of group
  if (bj0 + (JTILES - 1) < bi) return;              // group entirely below diag

  const int lane = threadIdx.x;                     // 0..31
  const int r16  = lane & 15;                       // row within 16-block
  const int kh   = lane >> 4;                       // K-half selector
  const int I    = bi  * 16;
  const int J0   = bj0 * 16;

  const int rowA = (I  + r16) * D_FEAT + 2 * kh;    // A-side gather base
  const int rowB = (J0 + r16) * D_FEAT + 2 * kh;    // B-side gather base (tile 0)

  v8f M1 [JTILES] = {};   // sum_d u[I+m]  * iv[J+n]   (tr+quad main, i->j)
  v8f M2 [JTILES] = {};   // sum_d mu[I+m] * mi[J+n]   (cross term,   i->j)
  v8f M1t[JTILES] = {};   // sum_d iv[I+m] * u[J+n]    (tr+quad main, j->i)
  v8f M2t[JTILES] = {};   // sum_d mi[I+m] * mu[J+n]   (cross term,   j->i)

#pragma unroll
  for (int ks = 0; ks < D_FEAT / 4; ++ks) {
    const int ka = rowA + 4 * ks;
    const v2f aU  = *(const v2f*)(u  + ka);   // A fragments: loaded once,
    const v2f aMu = *(const v2f*)(mu + ka);   // reused across all 4 J-tiles
    const v2f aIv = *(const v2f*)(iv + ka);
    const v2f aMi = *(const v2f*)(mi + ka);
#pragma unroll
    for (int t = 0; t < JTILES; ++t) {
      const int kb = rowB + t * (16 * D_FEAT) + 4 * ks;  // const-folded ioffset
      const v2f bIv = *(const v2f*)(iv + kb);
      const v2f bMi = *(const v2f*)(mi + kb);
      const v2f bU  = *(const v2f*)(u  + kb);
      const v2f bMu = *(const v2f*)(mu + kb);
      M1 [t] = __builtin_amdgcn_wmma_f32_16x16x4_f32(false, aU,  false, bIv, (short)0, M1 [t], false, false);
      M2 [t] = __builtin_amdgcn_wmma_f32_16x16x4_f32(false, aMu, false, bMi, (short)0, M2 [t], false, false);
      M1t[t] = __builtin_amdgcn_wmma_f32_16x16x4_f32(false, aIv, false, bU,  (short)0, M1t[t], false, false);
      M2t[t] = __builtin_amdgcn_wmma_f32_16x16x4_f32(false, aMi, false, bMu, (short)0, M2t[t], false, false);
    }
  }

  // C/D layout: VGPR r holds element (M = r + 8*kh, N = lane&15)
  float sI[8], pI[8]; int labI[8];
#pragma unroll
  for (int r = 0; r < 8; ++r) {
    const int i = I + r + 8 * kh;
    sI[r] = s[i]; pI[r] = p[i]; labI[r] = labels[i];
  }

  float pos = 0.0f, neg = 0.0f;
#pragma unroll
  for (int t = 0; t < JTILES; ++t) {
    const int bj = bj0 + t;
    // weight: below diag -> 0 (boundary waste), diag -> 1, above -> 2 (covers (J,I) too)
    const float w = (bj < bi) ? 0.0f : ((bj == bi) ? 1.0f : 2.0f);
    if (w == 0.0f) continue;                  // uniform across the wave

    const int   j    = bj * 16 + r16;         // column index, fixed per lane
    const float sJ   = s[j];
    const float pJ   = p[j];
    const int   labJ = labels[j];

    float tpos = 0.0f, tneg = 0.0f;
#pragma unroll
    for (int r = 0; r < 8; ++r) {
      const int   i  = I + r + 8 * kh;
      const float K1 = 0.5f * (M1 [t][r] - 2.0f * M2 [t][r] + sJ    + pJ    / (pI[r] + DET_EPS) - (float)D_FEAT);
      const float K2 = 0.5f * (M1t[t][r] - 2.0f * M2t[t][r] + sI[r] + pI[r] / (pJ    + DET_EPS) - (float)D_FEAT);
      const float sym   = 0.5f * (K1 + K2);
      const float logit = 1.0f / (1.0f + __expf(-sym));
      if (i != j) {
        if (labI[r] == labJ) tpos += logit; else tneg += logit;
      }
    }
    pos += w * tpos;
    neg += w * tneg;
  }

#pragma unroll
  for (int off = 16; off > 0; off >>= 1) {
    pos += __shfl_xor(pos, off, 32);
    neg += __shfl_xor(neg, off, 32);
  }
  if (lane == 0) {
    atomicAdd(&accum[0], pos);
    atomicAdd(&accum[1], neg);
  }
}

// ---------------- Phase 3: finalize -------------------------------------------
__global__ void kl_finalize_kernel(const float* __restrict__ accum,
                                   float* __restrict__ out) {
  if (threadIdx.x == 0 && blockIdx.x == 0) {
    const float denom = (float)N_BATCH * (float)N_BATCH;
    const float ps = accum[0];
    const float ns = accum[1];
    out[0] = ps / denom;   // mean(pos)
    out[1] = ns / denom;   // mean(neg)
    out[2] = ps;           // sum(pos)
    out[3] = ns;           // sum(neg)
  }
}

extern "C" void kernel_launch(void* const* d_in, const int* in_sizes, int n_in,
                              void* d_out, int out_size, void* d_ws, size_t ws_size,
                              hipStream_t stream) {
  const float* mu     = (const float*)d_in[0];
  const float* var    = (const float*)d_in[1];
  const int*   labels = (const int*)  d_in[2];

  float* ws    = (float*)d_ws;
  float* u     = ws;                               // [N*D]
  float* iv    = u  + N_BATCH * D_FEAT;            // [N*D]
  float* mi    = iv + N_BATCH * D_FEAT;            // [N*D]
  float* s     = mi + N_BATCH * D_FEAT;            // [N]
  float* p     = s  + N_BATCH;                     // [N]
  float* accum = p  + N_BATCH;                     // [2]

  kl_prep_kernel<<<N_BATCH, 32, 0, stream>>>(mu, var, u, iv, mi, s, p, accum);

  // bi = 0..255 ; J-groups of 4 tiles: 64 groups, 4 waves per block
  dim3 grid(NBLK, (NBLK / JTILES) / 4);
  dim3 block(32, 4);
  kl_tile_kernel<<<grid, block, 0, stream>>>(labels, mu, u, iv, mi, s, p, accum);

  kl_finalize_kernel<<<1, 32, 0, stream>>>(accum, (float*)d_out);
}